// SlotAttention_15479062135490
// MI455X (gfx1250) — compile-verified
//
#include <hip/hip_runtime.h>
#include <math.h>

#define B_   2
#define NS_  8
#define D_   64
#define HID_ 128
#define RES_ 128
#define HW_  (RES_*RES_)
#define EPS_ 1e-8f

typedef float v2f __attribute__((ext_vector_type(2)));
typedef float v8f __attribute__((ext_vector_type(8)));

// ---- WMMA fragment helpers (V_WMMA_F32_16X16X4_F32, wave32) ----
// A (16x4): lanes 0-15: M=lane, VGPR{0,1}=K{0,1}; lanes 16-31: M=lane-16, VGPR{0,1}=K{2,3}
__device__ __forceinline__ v2f fragA(const float* sm, int ld, int ks, int lane) {
  int m  = lane & 15;
  int kb = ks * 4 + ((lane < 16) ? 0 : 2);
  v2f a; a.x = sm[m * ld + kb]; a.y = sm[m * ld + kb + 1];
  return a;
}
// B (4x16) for X @ W^T with W row-major (out x in): B(k,n) = W[n0+n][k]
__device__ __forceinline__ v2f fragBW(const float* W, int ldw, int n0, int ks, int lane) {
  int n  = lane & 15;
  int kb = ks * 4 + ((lane < 16) ? 0 : 2);
  v2f b; b.x = W[(n0 + n) * ldw + kb]; b.y = W[(n0 + n) * ldw + kb + 1];
  return b;
}
__device__ __forceinline__ v8f wmma4(v2f a, v2f b, v8f c) {
  return __builtin_amdgcn_wmma_f32_16x16x4_f32(false, a, false, b, (short)0, c, false, false);
}

// ---- init slots/pos/scl ----
__global__ void k_init_state(const float* __restrict__ noise, const float* __restrict__ pl,
                             const float* __restrict__ sl, const float* __restrict__ mu,
                             const float* __restrict__ sg,
                             float* slots, float* pos, float* scl) {
  int tid = threadIdx.x;
  for (int idx = tid; idx < B_ * NS_ * D_; idx += 256) {
    int d = idx % D_;
    slots[idx] = mu[d] + sg[d] * noise[idx];
  }
  for (int idx = tid; idx < B_ * NS_ * 2; idx += 256) {
    int r = idx % (NS_ * 2);
    pos[idx] = pl[r];
    scl[idx] = sl[r];
  }
}

// ---- kx = LN(inputs)@Wk^T, vx = LN(inputs)@Wv^T  (WMMA f32) ----
__global__ void k_init_kv(const float* __restrict__ inp, const float* __restrict__ g,
                          const float* __restrict__ bb, const float* __restrict__ Wk,
                          const float* __restrict__ Wv, float* kx, float* vx) {
  __shared__ float sx[16 * 68];
  int lane = threadIdx.x;
  int base = blockIdx.x * 16;  // row base in [0, B*HW)
  if (lane < 16) {
    int m = lane;
    const float* row = inp + (size_t)(base + m) * D_;
    float mu = 0.f;
    for (int d = 0; d < D_; ++d) { float t = row[d]; sx[m * 68 + d] = t; mu += t; }
    mu *= (1.f / D_);
    float var = 0.f;
    for (int d = 0; d < D_; ++d) { float t = sx[m * 68 + d] - mu; var += t * t; }
    var *= (1.f / D_);
    float rstd = rsqrtf(var + 1e-5f);
    for (int d = 0; d < D_; ++d)
      sx[m * 68 + d] = (sx[m * 68 + d] - mu) * rstd * g[d] + bb[d];
  }
  __syncthreads();
  for (int w = 0; w < 2; ++w) {
    const float* W = w ? Wv : Wk;
    float* out = w ? vx : kx;
    for (int nt = 0; nt < 4; ++nt) {
      v8f acc = {0.f,0.f,0.f,0.f,0.f,0.f,0.f,0.f};
      #pragma unroll
      for (int ks = 0; ks < 16; ++ks)
        acc = wmma4(fragA(sx, 68, ks, lane), fragBW(W, 64, nt * 16, ks, lane), acc);
      int n  = nt * 16 + (lane & 15);
      int mb = (lane < 16) ? 0 : 8;
      #pragma unroll
      for (int v = 0; v < 8; ++v)
        out[(size_t)(base + mb + v) * D_ + n] = acc[v];
    }
  }
}

// ---- q = LN(slots,nsl)@Wq^T ----
__global__ void k_q(const float* __restrict__ slots, const float* __restrict__ g,
                    const float* __restrict__ bb, const float* __restrict__ Wq, float* q) {
  int tid = threadIdx.x;           // 256 threads: 16 rows x 16 col-groups
  int row = tid >> 4;
  int c0  = (tid & 15) * 4;
  const float* s = slots + row * D_;
  float mu = 0.f;
  for (int d = 0; d < D_; ++d) mu += s[d];
  mu *= (1.f / D_);
  float var = 0.f;
  for (int d = 0; d < D_; ++d) { float t = s[d] - mu; var += t * t; }
  var *= (1.f / D_);
  float rstd = rsqrtf(var + 1e-5f);
  for (int cc = 0; cc < 4; ++cc) {
    int c = c0 + cc;
    float acc = 0.f;
    for (int k = 0; k < D_; ++k)
      acc += ((s[k] - mu) * rstd * g[k] + bb[k]) * Wq[c * D_ + k];
    q[row * D_ + c] = acc;
  }
}

// ---- zero accumulators ----
__global__ void k_zero(float* upd, float* S0, float* S1, float* S2) {
  int tid = threadIdx.x;
  for (int i = tid; i < B_ * NS_ * D_; i += 256) upd[i] = 0.f;
  if (tid < B_ * NS_) S0[tid] = 0.f;
  if (tid < B_ * NS_ * 2) { S1[tid] = 0.f; S2[tid] = 0.f; }
}

// ---- fused main: ge -> LN -> MLP(k,v) [WMMA] -> dots -> softmax(i) -> accumulators ----
__global__ void k_main(const float* __restrict__ kx, const float* __restrict__ vx,
                       const float* __restrict__ q,  const float* __restrict__ pos,
                       const float* __restrict__ scl,
                       const float* __restrict__ Wg, const float* __restrict__ bg,
                       const float* __restrict__ lng, const float* __restrict__ lnb,
                       const float* __restrict__ Wm1, const float* __restrict__ bm1,
                       const float* __restrict__ Wm2, const float* __restrict__ bm2,
                       float* upd, float* S0, float* S1, float* S2) {
  __shared__ float sq[NS_ * D_];
  __shared__ float sg0[16], sg1[16], sr0[16], sr1[16];
  __shared__ float st[16 * 68];
  __shared__ float sh[16 * 132];
  __shared__ float skk[16 * 68];
  __shared__ float svv[16 * 68];
  __shared__ float sdots[NS_ * 16];
  __shared__ float sattn[NS_ * 16];

  int lane = threadIdx.x;
  int tile = blockIdx.x, j = blockIdx.y, b = blockIdx.z;

  for (int idx = lane; idx < NS_ * D_; idx += 32) sq[idx] = q[b * NS_ * D_ + idx];
  if (lane < 16) {
    int p0 = tile * 16 + lane;
    int row = p0 >> 7, col = p0 & 127;
    float g0 = -1.f + 2.f * row / (RES_ - 1.f);
    float g1 = -1.f + 2.f * col / (RES_ - 1.f);
    sg0[lane] = g0; sg1[lane] = g1;
    float px = pos[(b * NS_ + j) * 2 + 0], py = pos[(b * NS_ + j) * 2 + 1];
    float sx = scl[(b * NS_ + j) * 2 + 0] * 5.f + EPS_;
    float sy = scl[(b * NS_ + j) * 2 + 1] * 5.f + EPS_;
    sr0[lane] = (g0 - px) / sx;
    sr1[lane] = (g1 - py) / sy;
  }
  __syncthreads();

  size_t rowbase = (size_t)b * HW_ + (size_t)tile * 16;
  for (int ph = 0; ph < 2; ++ph) {
    const float* src = ph ? vx : kx;
    float* outsm = ph ? svv : skk;
    if (lane < 16) {                         // t = src + ge ; LayerNorm(mlp_ln)
      int m = lane;
      const float* row = src + (rowbase + m) * D_;
      float r0 = sr0[m], r1 = sr1[m];
      float mu = 0.f;
      for (int d = 0; d < D_; ++d) {
        float t = row[d] + r0 * Wg[d * 2] + r1 * Wg[d * 2 + 1] + bg[d];
        st[m * 68 + d] = t; mu += t;
      }
      mu *= (1.f / D_);
      float var = 0.f;
      for (int d = 0; d < D_; ++d) { float t = st[m * 68 + d] - mu; var += t * t; }
      var *= (1.f / D_);
      float rstd = rsqrtf(var + 1e-5f);
      for (int d = 0; d < D_; ++d)
        st[m * 68 + d] = (st[m * 68 + d] - mu) * rstd * lng[d] + lnb[d];
    }
    __syncthreads();
    // GEMM1: h = relu(st @ Wm1^T + bm1), 16x128, K=64
    for (int nt = 0; nt < 8; ++nt) {
      v8f acc = {0.f,0.f,0.f,0.f,0.f,0.f,0.f,0.f};
      #pragma unroll
      for (int ks = 0; ks < 16; ++ks)
        acc = wmma4(fragA(st, 68, ks, lane), fragBW(Wm1, 64, nt * 16, ks, lane), acc);
      int n  = nt * 16 + (lane & 15);
      int mb = (lane < 16) ? 0 : 8;
      float bn = bm1[n];
      #pragma unroll
      for (int v = 0; v < 8; ++v) {
        float h = acc[v] + bn;
        sh[(mb + v) * 132 + n] = h > 0.f ? h : 0.f;
      }
    }
    __syncthreads();
    // GEMM2: out = h @ Wm2^T + bm2, 16x64, K=128
    for (int nt = 0; nt < 4; ++nt) {
      v8f acc = {0.f,0.f,0.f,0.f,0.f,0.f,0.f,0.f};
      #pragma unroll
      for (int ks = 0; ks < 32; ++ks)
        acc = wmma4(fragA(sh, 132, ks, lane), fragBW(Wm2, 128, nt * 16, ks, lane), acc);
      int n  = nt * 16 + (lane & 15);
      int mb = (lane < 16) ? 0 : 8;
      float bn = bm2[n];
      #pragma unroll
      for (int v = 0; v < 8; ++v)
        outsm[(mb + v) * 68 + n] = acc[v] + bn;
    }
    __syncthreads();
  }

  // dots[i][m] = scale * q_i . k_m   (8x16 pairs, 4 per lane)
  for (int t = 0; t < 4; ++t) {
    int p = lane + 32 * t;
    int i = p >> 4, m = p & 15;
    float s = 0.f;
    for (int d = 0; d < D_; ++d) s += sq[i * D_ + d] * skk[m * 68 + d];
    sdots[i * 16 + m] = s * 0.125f;  // D^-0.5
  }
  __syncthreads();
  // softmax over i (query slots) per position m, + eps
  if (lane < 16) {
    int m = lane;
    float mx = sdots[m];
    #pragma unroll
    for (int i = 1; i < NS_; ++i) mx = fmaxf(mx, sdots[i * 16 + m]);
    float e[NS_]; float sum = 0.f;
    #pragma unroll
    for (int i = 0; i < NS_; ++i) { e[i] = __expf(sdots[i * 16 + m] - mx); sum += e[i]; }
    float inv = 1.f / sum;
    #pragma unroll
    for (int i = 0; i < NS_; ++i) sattn[i * 16 + m] = e[i] * inv + EPS_;
  }
  __syncthreads();
  // updates[b,i,:] += attn^T @ v  (atomic across (j,tile) blocks)
  {
    int i = lane >> 2, d0 = lane & 3;
    for (int t = 0; t < 16; ++t) {
      int d = d0 + 4 * t;
      float s = 0.f;
      #pragma unroll
      for (int m = 0; m < 16; ++m) s += sattn[i * 16 + m] * svv[m * 68 + d];
      atomicAdd(&upd[(b * NS_ + i) * D_ + d], s);
    }
  }
  // moment accumulators for pos/scl
  if (lane < NS_) {
    int i = lane;
    float s0 = 0.f, s10 = 0.f, s11 = 0.f, s20 = 0.f, s21 = 0.f;
    #pragma unroll
    for (int m = 0; m < 16; ++m) {
      float a = sattn[i * 16 + m];
      s0 += a;
      s10 += a * sg0[m]; s11 += a * sg1[m];
      s20 += a * sg0[m] * sg0[m]; s21 += a * sg1[m] * sg1[m];
    }
    atomicAdd(&S0[b * NS_ + i], s0);
    atomicAdd(&S1[(b * NS_ + i) * 2 + 0], s10);
    atomicAdd(&S1[(b * NS_ + i) * 2 + 1], s11);
    atomicAdd(&S2[(b * NS_ + i) * 2 + 0], s20);
    atomicAdd(&S2[(b * NS_ + i) * 2 + 1], s21);
  }
}

// ---- GRU + residual MLP slot update (16 rows, tiny) ----
__global__ void k_gru(const float* __restrict__ upd, float* slots,
                      const float* __restrict__ wih, const float* __restrict__ whh,
                      const float* __restrict__ bih, const float* __restrict__ bhh,
                      const float* __restrict__ npg, const float* __restrict__ npb,
                      const float* __restrict__ f1w, const float* __restrict__ f1b,
                      const float* __restrict__ f2w, const float* __restrict__ f2b) {
  __shared__ float gi[16 * 192];
  __shared__ float gh[16 * 192];
  __shared__ float sn[16 * 64];
  __shared__ float smu[16], srs[16];
  __shared__ float h1[16 * 128];
  int tid = threadIdx.x;
  for (int idx = tid; idx < 16 * 192; idx += 256) {
    int r = idx / 192, c = idx % 192;
    float a = bih[c], h = bhh[c];
    for (int k = 0; k < D_; ++k) {
      a += upd[r * D_ + k] * wih[c * D_ + k];
      h += slots[r * D_ + k] * whh[c * D_ + k];
    }
    gi[idx] = a; gh[idx] = h;
  }
  __syncthreads();
  for (int idx = tid; idx < 16 * 64; idx += 256) {
    int r = idx / 64, d = idx % 64;
    float rg = 1.f / (1.f + __expf(-(gi[r * 192 + d] + gh[r * 192 + d])));
    float z  = 1.f / (1.f + __expf(-(gi[r * 192 + 64 + d] + gh[r * 192 + 64 + d])));
    float nn = tanhf(gi[r * 192 + 128 + d] + rg * gh[r * 192 + 128 + d]);
    sn[idx] = (1.f - z) * nn + z * slots[r * D_ + d];
  }
  __syncthreads();
  if (tid < 16) {
    float mu = 0.f;
    for (int d = 0; d < 64; ++d) mu += sn[tid * 64 + d];
    mu *= (1.f / 64.f);
    float var = 0.f;
    for (int d = 0; d < 64; ++d) { float t = sn[tid * 64 + d] - mu; var += t * t; }
    var *= (1.f / 64.f);
    smu[tid] = mu; srs[tid] = rsqrtf(var + 1e-5f);
  }
  __syncthreads();
  for (int idx = tid; idx < 16 * 128; idx += 256) {
    int r = idx / 128, c = idx % 128;
    float acc = f1b[c];
    for (int k = 0; k < 64; ++k)
      acc += ((sn[r * 64 + k] - smu[r]) * srs[r] * npg[k] + npb[k]) * f1w[c * 64 + k];
    h1[idx] = acc > 0.f ? acc : 0.f;
  }
  __syncthreads();
  for (int idx = tid; idx < 16 * 64; idx += 256) {
    int r = idx / 64, d = idx % 64;
    float acc = f2b[d];
    for (int c = 0; c < 128; ++c) acc += h1[r * 128 + c] * f2w[d * 128 + c];
    slots[idx] = sn[idx] + acc;
  }
}

// ---- finalize pos & scl from moments (closed-form incl. eps term) ----
__global__ void k_fin(const float* __restrict__ S0, const float* __restrict__ S1,
                      const float* __restrict__ S2, float* pos, float* scl) {
  int tid = threadIdx.x;
  if (tid >= B_ * NS_) return;
  float sumr = 0.f, sumr2 = 0.f;
  for (int t = 0; t < RES_; ++t) {
    float r = -1.f + 2.f * t / (RES_ - 1.f);
    sumr += r; sumr2 += r * r;
  }
  float G1 = RES_ * sumr, G2 = RES_ * sumr2;   // per-channel grid sums
  float T = S0[tid];
  for (int l = 0; l < 2; ++l) {
    float s1 = S1[tid * 2 + l], s2 = S2[tid * 2 + l];
    float p = s1 / T;
    float a = (s2 - 2.f * p * s1 + p * p * T) / T;               // sum attn_n*(g-p)^2
    float bterm = EPS_ * (float)NS_ * (G2 - 2.f * p * G1 + (float)HW_ * p * p);
    pos[tid * 2 + l] = p;
    scl[tid * 2 + l] = sqrtf(a + bterm);
  }
}

// ---- pack outputs: slots(1024), pos(32), scl(32) ----
__global__ void k_out(const float* __restrict__ slots, const float* __restrict__ pos,
                      const float* __restrict__ scl, float* out) {
  int tid = threadIdx.x;
  for (int i = tid; i < B_ * NS_ * D_; i += 256) out[i] = slots[i];
  if (tid < 32) out[1024 + tid] = pos[tid];
  if (tid < 32) out[1056 + tid] = scl[tid];
}

extern "C" void kernel_launch(void* const* d_in, const int* in_sizes, int n_in,
                              void* d_out, int out_size, void* d_ws, size_t ws_size,
                              hipStream_t stream) {
  (void)in_sizes; (void)n_in; (void)out_size; (void)ws_size;
  const float* inputs = (const float*)d_in[0];
  const float* noise  = (const float*)d_in[1];
  const float* p_lat  = (const float*)d_in[2];
  const float* s_lat  = (const float*)d_in[3];
  const float* s_mu   = (const float*)d_in[4];
  const float* s_sg   = (const float*)d_in[5];
  const float* Wq     = (const float*)d_in[6];
  const float* Wg     = (const float*)d_in[7];
  const float* bg     = (const float*)d_in[8];
  const float* Wk     = (const float*)d_in[9];
  const float* Wv     = (const float*)d_in[10];
  const float* lng    = (const float*)d_in[11];
  const float* lnb    = (const float*)d_in[12];
  const float* Wm1    = (const float*)d_in[13];
  const float* bm1    = (const float*)d_in[14];
  const float* Wm2    = (const float*)d_in[15];
  const float* bm2    = (const float*)d_in[16];
  const float* wih    = (const float*)d_in[17];
  const float* whh    = (const float*)d_in[18];
  const float* bih    = (const float*)d_in[19];
  const float* bhh    = (const float*)d_in[20];
  const float* f1w    = (const float*)d_in[21];
  const float* f1b    = (const float*)d_in[22];
  const float* f2w    = (const float*)d_in[23];
  const float* f2b    = (const float*)d_in[24];
  const float* nin_g  = (const float*)d_in[25];
  const float* nin_b  = (const float*)d_in[26];
  const float* nsl_g  = (const float*)d_in[27];
  const float* nsl_b  = (const float*)d_in[28];
  const float* npf_g  = (const float*)d_in[29];
  const float* npf_b  = (const float*)d_in[30];

  float* ws    = (float*)d_ws;
  float* kx    = ws;                    // B*HW*D
  float* vx    = kx + (size_t)B_ * HW_ * D_;
  float* slots = vx + (size_t)B_ * HW_ * D_;
  float* q     = slots + B_ * NS_ * D_;
  float* pos   = q + B_ * NS_ * D_;
  float* scl   = pos + B_ * NS_ * 2;
  float* upd   = scl + B_ * NS_ * 2;
  float* S0    = upd + B_ * NS_ * D_;
  float* S1    = S0 + B_ * NS_;
  float* S2    = S1 + B_ * NS_ * 2;

  k_init_state<<<1, 256, 0, stream>>>(noise, p_lat, s_lat, s_mu, s_sg, slots, pos, scl);
  k_init_kv<<<(B_ * HW_) / 16, 32, 0, stream>>>(inputs, nin_g, nin_b, Wk, Wv, kx, vx);

  for (int it = 0; it < 3; ++it) {
    k_zero<<<1, 256, 0, stream>>>(upd, S0, S1, S2);
    k_q<<<1, 256, 0, stream>>>(slots, nsl_g, nsl_b, Wq, q);
    k_main<<<dim3(HW_ / 16, NS_, B_), 32, 0, stream>>>(
        kx, vx, q, pos, scl, Wg, bg, lng, lnb, Wm1, bm1, Wm2, bm2, upd, S0, S1, S2);
    if (it != 2)
      k_gru<<<1, 256, 0, stream>>>(upd, slots, wih, whh, bih, bhh,
                                   npf_g, npf_b, f1w, f1b, f2w, f2b);
    k_fin<<<1, 32, 0, stream>>>(S0, S1, S2, pos, scl);
  }
  k_out<<<1, 256, 0, stream>>>(slots, pos, scl, (float*)d_out);
}